// BILINEAR_56169582297414
// MI455X (gfx1250) — compile-verified
//
#include <hip/hip_runtime.h>
#include <hip/hip_bf16.h>

typedef float v2f __attribute__((ext_vector_type(2)));
typedef float v8f __attribute__((ext_vector_type(8)));

// D = A(16x4) x B(4x16) + C, fp32 exact.
__device__ __forceinline__ v8f wmma_f32_4(v2f a, v2f b, v8f c) {
    return __builtin_amdgcn_wmma_f32_16x16x4_f32(
        /*neg_a=*/false, a, /*neg_b=*/false, b,
        /*c_mod=*/(short)0, c, /*reuse_a=*/false, /*reuse_b=*/false);
}

#if defined(__has_builtin)
#if __has_builtin(__builtin_amdgcn_permlane16)
#define HAVE_PERMLANE16 1
#endif
#endif
#ifndef HAVE_PERMLANE16
#define HAVE_PERMLANE16 0
#endif

// Lane-xor exchange within 16-lane rows: single VALU v_permlane16_b32, no LDS traffic.
// Selector nibble i = i ^ mask (within-row lane index).
#if HAVE_PERMLANE16
#define LANE_XOR(x, S0, S1)                                                          \
    __uint_as_float(__builtin_amdgcn_permlane16(__float_as_uint(x),                  \
        __float_as_uint(x), (unsigned)(S0), (unsigned)(S1), false, false))
__device__ __forceinline__ float lane_xor1(float x) { return LANE_XOR(x, 0x67452301u, 0xEFCDAB89u); }
__device__ __forceinline__ float lane_xor2(float x) { return LANE_XOR(x, 0x54761032u, 0xDCFE98BAu); }
__device__ __forceinline__ float lane_xor4(float x) { return LANE_XOR(x, 0x32107654u, 0xBA98FEDCu); }
__device__ __forceinline__ float lane_xor8(float x) { return LANE_XOR(x, 0xFEDCBA98u, 0x76543210u); }
#else
__device__ __forceinline__ float lane_xor1(float x) { return __shfl_xor(x, 1); }
__device__ __forceinline__ float lane_xor2(float x) { return __shfl_xor(x, 2); }
__device__ __forceinline__ float lane_xor4(float x) { return __shfl_xor(x, 4); }
__device__ __forceinline__ float lane_xor8(float x) { return __shfl_xor(x, 8); }
#endif

// One wave handles a tile of 16 groups. 8 waves per block -> 128 groups / block.
__global__ __launch_bounds__(256) void BILINEAR_group_kernel(
    const int* __restrict__ item_inputs,          // [B]
    const int* __restrict__ member_ids,           // [B,16]
    const unsigned char* __restrict__ member_mask,// [B,16] bool
    const float* __restrict__ user_table,         // [NU,32]
    const float* __restrict__ item_table,         // [NI,32]
    const float* __restrict__ W_bil,              // [32,32]
    const float* __restrict__ b_bil,              // [1]
    const float* __restrict__ W1,                 // [8,96]
    const float* __restrict__ b1,                 // [8]
    const float* __restrict__ W2,                 // [1,8]
    const float* __restrict__ b2,                 // [1]
    float* __restrict__ out,                      // [B]
    int B)
{
    __shared__ __align__(16) float V_lds[8 * 512];   // per-wave V[16][32]
    __shared__ __align__(16) float FU_lds[8 * 512];  // per-wave final_user[16][32]

    const int lane = threadIdx.x & 31;
    const int wv   = threadIdx.x >> 5;
    const int gb0  = (blockIdx.x * 8 + wv) * 16;
    if (gb0 >= B) return;                            // B % 128 == 0 in practice

    const int half = lane >> 4;   // 0: lanes 0-15, 1: lanes 16-31
    const int lm   = lane & 15;

    // ---------------- Phase 1: V[b,d] = sum_e IE[b,e] * W_bil[d,e] via WMMA ----
    const int ii = item_inputs[gb0 + lm];
    const float* itrow = item_table + (size_t)ii * 32;

    v8f acc0 = {};   // d = 0..15
    v8f acc1 = {};   // d = 16..31
    v2f ia[8];       // item-embedding A fragments, reused for the MLP GEMM
#pragma unroll
    for (int kk = 0; kk < 8; ++kk) {
        const int k0 = kk * 4 + half * 2;
        v2f a = { itrow[k0], itrow[k0 + 1] };
        ia[kk] = a;
        const float* wp0 = W_bil + lm * 32 + k0;          // B[e, d=lm]     = W[d,e]
        const float* wp1 = W_bil + (lm + 16) * 32 + k0;   // B[e, d=lm+16]
        v2f bv0 = { wp0[0], wp0[1] };
        v2f bv1 = { wp1[0], wp1[1] };
        acc0 = wmma_f32_4(a, bv0, acc0);
        acc1 = wmma_f32_4(a, bv1, acc1);
    }
#pragma unroll
    for (int r = 0; r < 8; ++r) {   // D: VGPR r, lane L -> V[b=r+8*half, d=lm(+16)]
        V_lds[wv * 512 + (r + 8 * half) * 32 + lm]      = acc0[r];
        V_lds[wv * 512 + (r + 8 * half) * 32 + lm + 16] = acc1[r];
    }
    __syncthreads();

    // ---------------- Phase 2: scores + masked weighted member sum -------------
    // 2 groups per pass: lanes 0-15 -> b=2p (member m=lm), lanes 16-31 -> b=2p+1.
    const float bbil = b_bil[0];
    const bool l0 = (lm & 1) != 0, l1 = (lm & 2) != 0,
               l2 = (lm & 4) != 0, l3 = (lm & 8) != 0;
    // final d-offset this lane owns after reduce-scatter (2 consecutive d's)
    const int dbase = ((lm & 1) << 4) | ((lm & 2) << 2) | (lm & 4) | ((lm & 8) >> 2);

    for (int p = 0; p < 8; ++p) {
        const int bl  = 2 * p + half;
        const int gbi = gb0 + bl;
        const int mid   = member_ids[(size_t)gbi * 16 + lm];          // coalesced 128B
        const float msk = (float)member_mask[(size_t)gbi * 16 + lm];  // coalesced 32B

        const float4* ur = (const float4*)(user_table + (size_t)mid * 32); // 1 line/row
        float4 r4[8];
#pragma unroll
        for (int q = 0; q < 8; ++q) r4[q] = ur[q];

        const float4* vr = (const float4*)(V_lds + wv * 512 + bl * 32);
        float sc = 0.0f;
#pragma unroll
        for (int q = 0; q < 8; ++q) {
            float4 vq = vr[q];
            sc += r4[q].x * vq.x + r4[q].y * vq.y + r4[q].z * vq.z + r4[q].w * vq.w;
        }
        const float w = (sc + bbil) * msk;   // bias then mask, as in reference

        float v0[32];
#pragma unroll
        for (int q = 0; q < 8; ++q) {
            v0[q * 4 + 0] = r4[q].x * w;
            v0[q * 4 + 1] = r4[q].y * w;
            v0[q * 4 + 2] = r4[q].z * w;
            v0[q * 4 + 3] = r4[q].w * w;
        }

        // Reduce-scatter butterfly over the 16 member-lanes of each half:
        // live set halves each step (32 -> 16 -> 8 -> 4 -> 2 values per lane).
        float s1[16];
#pragma unroll
        for (int j = 0; j < 16; ++j) {          // xor 1: keep d with bit4 == l0
            float pa = lane_xor1(v0[j]);
            float pb = lane_xor1(v0[j + 16]);
            s1[j] = (l0 ? v0[j + 16] : v0[j]) + (l0 ? pb : pa);
        }
        float s2[8];
#pragma unroll
        for (int j = 0; j < 8; ++j) {           // xor 2: keep d with bit3 == l1
            float pa = lane_xor2(s1[j]);
            float pb = lane_xor2(s1[j + 8]);
            s2[j] = (l1 ? s1[j + 8] : s1[j]) + (l1 ? pb : pa);
        }
        float s3[4];
#pragma unroll
        for (int j = 0; j < 4; ++j) {           // xor 4: keep d with bit2 == l2
            float pa = lane_xor4(s2[j]);
            float pb = lane_xor4(s2[j + 4]);
            s3[j] = (l2 ? s2[j + 4] : s2[j]) + (l2 ? pb : pa);
        }
        float s4[2];
#pragma unroll
        for (int j = 0; j < 2; ++j) {           // xor 8: keep d with bit1 == l3
            float pa = lane_xor8(s3[j]);
            float pb = lane_xor8(s3[j + 2]);
            s4[j] = (l3 ? s3[j + 2] : s3[j]) + (l3 ? pb : pa);
        }
        // lane owns fu[bl][dbase], fu[bl][dbase+1] -> one b64 store, all lanes active
        float2 fu2; fu2.x = s4[0]; fu2.y = s4[1];
        *(float2*)(FU_lds + wv * 512 + bl * 32 + dbase) = fu2;
    }
    __syncthreads();

    // ---------------- Phase 3: H = relu(NE @ W1^T + b1) via WMMA ---------------
    // NE[b,0:32)=fu*ie, [32:64)=fu, [64:96)=ie. A fragments are elementwise in (b,k),
    // so the fu*ie fragment is just fu_frag * ie_frag.
    v8f hacc = {};
    v2f ffu[8];
#pragma unroll
    for (int kk = 0; kk < 8; ++kk) {             // k = 0..31 : fu*ie
        const int k0 = kk * 4 + half * 2;
        const float* fp = FU_lds + wv * 512 + lm * 32 + k0;
        v2f fu2 = { fp[0], fp[1] };
        ffu[kk] = fu2;
        v2f a = fu2 * ia[kk];
        v2f bm = { 0.0f, 0.0f };
        if (lm < 8) { const float* wp = W1 + lm * 96 + k0;      bm = (v2f){ wp[0], wp[1] }; }
        hacc = wmma_f32_4(a, bm, hacc);
    }
#pragma unroll
    for (int kk = 0; kk < 8; ++kk) {             // k = 32..63 : fu
        const int k0 = kk * 4 + half * 2;
        v2f bm = { 0.0f, 0.0f };
        if (lm < 8) { const float* wp = W1 + lm * 96 + 32 + k0; bm = (v2f){ wp[0], wp[1] }; }
        hacc = wmma_f32_4(ffu[kk], bm, hacc);
    }
#pragma unroll
    for (int kk = 0; kk < 8; ++kk) {             // k = 64..95 : ie
        const int k0 = kk * 4 + half * 2;
        v2f bm = { 0.0f, 0.0f };
        if (lm < 8) { const float* wp = W1 + lm * 96 + 64 + k0; bm = (v2f){ wp[0], wp[1] }; }
        hacc = wmma_f32_4(ia[kk], bm, hacc);
    }

    // ---------------- Phase 4: bias, relu, W2 dot, sigmoid, store --------------
    const float bb1 = (lm < 8) ? b1[lm] : 0.0f;
    const float w2v = (lm < 8) ? W2[lm] : 0.0f;
    const float bb2 = b2[0];
#pragma unroll
    for (int r = 0; r < 8; ++r) {
        float h = fmaxf(hacc[r] + bb1, 0.0f);    // cols 8..15 are exactly 0
        float t = h * w2v;
        t += lane_xor1(t);
        t += lane_xor2(t);
        t += lane_xor4(t);
        t += lane_xor8(t);
        if (lm == r) {
            const float y = 1.0f / (1.0f + __expf(-(t + bb2)));
            out[gb0 + r + 8 * half] = y;
        }
    }
}

extern "C" void kernel_launch(void* const* d_in, const int* in_sizes, int n_in,
                              void* d_out, int out_size, void* d_ws, size_t ws_size,
                              hipStream_t stream) {
    const int*           item_inputs = (const int*)d_in[0];
    const int*           member_ids  = (const int*)d_in[1];
    const unsigned char* member_mask = (const unsigned char*)d_in[2];
    const float*         user_table  = (const float*)d_in[3];
    const float*         item_table  = (const float*)d_in[4];
    const float*         W_bil       = (const float*)d_in[5];
    const float*         b_bil       = (const float*)d_in[6];
    const float*         W1          = (const float*)d_in[7];
    const float*         b1          = (const float*)d_in[8];
    const float*         W2          = (const float*)d_in[9];
    const float*         b2          = (const float*)d_in[10];
    float* out = (float*)d_out;

    const int B = in_sizes[0];
    const int blocks = (B + 127) / 128;   // 8 waves/block, 16 groups/wave
    BILINEAR_group_kernel<<<blocks, 256, 0, stream>>>(
        item_inputs, member_ids, member_mask, user_table, item_table,
        W_bil, b_bil, W1, b1, W2, b2, out, B);
}